// MultiHeadAttention_3332894621781
// MI455X (gfx1250) — compile-verified
//
#include <hip/hip_runtime.h>
#include <hip/hip_bf16.h>

typedef __attribute__((ext_vector_type(2))) float v2f;
typedef __attribute__((ext_vector_type(8))) float v8f;

#define D_MODEL 1024
#define N_HEAD  16
#define D_HEAD  64
#define BATCH   2
#define SEQ     2048

static __device__ __forceinline__ v8f wmma_f32(v2f a, v2f b, v8f c) {
  // D = A(16x4,f32) * B(4x16,f32) + C(16x16,f32)
  return __builtin_amdgcn_wmma_f32_16x16x4_f32(false, a, false, b, (short)0, c,
                                               false, false);
}

// ---------------------------------------------------------------------------
// Generic GEMM + bias: Y[M,N] = X[M,K] @ W[K,N] + bias[N]
// Block tile 64x128, K staged in 32-wide LDS chunks. 8 waves; wave w owns the
// 64x16 column strip n = n0 + 16*w as four stacked 16x16 WMMA accumulators.
// ---------------------------------------------------------------------------
#define GBM 64
#define GBN 128
#define GBK 32
#define LA_STRIDE 34   // 64x32 A tile, +2 pad: conflict-free 16-row f32x2 reads
#define LB_STRIDE 136  // 32x128 B tile, +8 pad: keeps float4 stores 16B-aligned

__global__ __launch_bounds__(256) void gemm_bias_wmma(
    const float* __restrict__ X, const float* __restrict__ W,
    const float* __restrict__ bias, float* __restrict__ Y,
    int M, int N, int K) {
  __shared__ float lA[GBM * LA_STRIDE];
  __shared__ float lB[GBK * LB_STRIDE];

  const int tid  = threadIdx.x;
  const int lane = tid & 31;
  const int wave = tid >> 5;
  const int hi   = lane >> 4;  // lane half: selects K pair / M+8
  const int lo   = lane & 15;
  const int m0   = blockIdx.y * GBM;
  const int n0   = blockIdx.x * GBN;

  v8f acc[4];
  for (int i = 0; i < 4; ++i)
    for (int e = 0; e < 8; ++e) acc[i][e] = 0.0f;

  for (int k0 = 0; k0 < K; k0 += GBK) {
    // A tile 64x32: 1024 float2 loads, 4 per thread
    for (int j = 0; j < 4; ++j) {
      int idx = tid + 256 * j;
      int row = idx >> 4, c2 = (idx & 15) << 1;
      float2 t = *reinterpret_cast<const float2*>(
          X + (size_t)(m0 + row) * K + k0 + c2);
      *reinterpret_cast<float2*>(&lA[row * LA_STRIDE + c2]) = t;
    }
    // B tile 32x128: 1024 float4 loads, 4 per thread
    for (int j = 0; j < 4; ++j) {
      int idx = tid + 256 * j;
      int row = idx >> 5, c4 = (idx & 31) << 2;
      float4 t = *reinterpret_cast<const float4*>(
          W + (size_t)(k0 + row) * N + n0 + c4);
      *reinterpret_cast<float4*>(&lB[row * LB_STRIDE + c4]) = t;
    }
    __syncthreads();

    for (int kk = 0; kk < GBK; kk += 4) {
      const int kr = kk + 2 * hi;
      v2f bf;
      bf.x = lB[kr * LB_STRIDE + wave * 16 + lo];
      bf.y = lB[(kr + 1) * LB_STRIDE + wave * 16 + lo];
      for (int i = 0; i < 4; ++i) {
        const int ar = i * 16 + lo;
        v2f af = *reinterpret_cast<const v2f*>(&lA[ar * LA_STRIDE + kr]);
        acc[i] = wmma_f32(af, bf, acc[i]);
      }
    }
    __syncthreads();
  }

  const int col = n0 + wave * 16 + lo;
  const float bb = bias[col];
  for (int i = 0; i < 4; ++i)
    for (int r = 0; r < 8; ++r) {
      const int row = m0 + i * 16 + r + 8 * hi;
      Y[(size_t)row * N + col] = acc[i][r] + bb;
    }
}

// ---------------------------------------------------------------------------
// Causal attention for one (batch, head, 16-row query tile).
// Phase A: scores[16 x 2048] via WMMA into dynamic LDS (causal tiles skipped).
// Phase B: row softmax (shfl reductions), probs written to LDS + global attn.
// Phase C: ctx[16 x 64] = P @ V via WMMA, A-fragments read from LDS.
// ---------------------------------------------------------------------------
#define SC_STRIDE 2052  // 2048 + 4 pad: decorrelates LDS banks across rows

__global__ __launch_bounds__(256) void attn_softmax_wmma(
    const float* __restrict__ Q, const float* __restrict__ Km,
    const float* __restrict__ V, float* __restrict__ attn,
    float* __restrict__ ctx) {
  extern __shared__ float sc[];  // [16][SC_STRIDE]

  const int tid  = threadIdx.x;
  const int lane = tid & 31;
  const int wave = tid >> 5;
  const int hi   = lane >> 4;
  const int lo   = lane & 15;
  const int qt   = blockIdx.x;       // query tile, 0..127
  const int bh   = blockIdx.y;       // 0..31
  const int b    = bh >> 4;
  const int h    = bh & 15;
  const int q0   = qt * 16;
  const int tokBase = b * SEQ;
  const int hOff    = h * D_HEAD;

  // ---- Phase A: scores = (Q K^T) / sqrt(Dh), causal-masked -> LDS
  for (int it = 0; it < SEQ / (16 * 8); ++it) {
    const int n0 = (it * 8 + wave) * 16;
    if (n0 > q0 + 15) {  // fully masked key tile: exp() -> 0 later
      for (int r = 0; r < 8; ++r)
        sc[(r + 8 * hi) * SC_STRIDE + n0 + lo] = -1e9f;
    } else {
      v8f acc;
      for (int e = 0; e < 8; ++e) acc[e] = 0.0f;
      const float* qrow = Q + (size_t)(tokBase + q0 + lo) * D_MODEL + hOff;
      const float* krow = Km + (size_t)(tokBase + n0 + lo) * D_MODEL + hOff;
      for (int k = 0; k < D_HEAD; k += 4) {
        const int kr = k + 2 * hi;
        v2f af = *reinterpret_cast<const v2f*>(qrow + kr);  // A[m=lo][kr..kr+1]
        v2f bf = *reinterpret_cast<const v2f*>(krow + kr);  // B[kr..][n=lo] = K[n][kr..]
        acc = wmma_f32(af, bf, acc);
      }
      for (int r = 0; r < 8; ++r) {
        const int m = r + 8 * hi;
        const int n = n0 + lo;
        float s = acc[r] * 0.125f;  // 1/sqrt(64)
        if (n > q0 + m) s = -1e9f;
        sc[m * SC_STRIDE + n] = s;
      }
    }
  }
  __syncthreads();

  // ---- Phase B: softmax over each of 16 rows; 2 rows per wave
  for (int rr = 0; rr < 2; ++rr) {
    const int m  = wave * 2 + rr;
    const int qg = q0 + m;
    float* row = sc + m * SC_STRIDE;
    float mx = -3.4e38f;
    for (int n = lane; n < SEQ; n += 32) mx = fmaxf(mx, row[n]);
    for (int off = 16; off > 0; off >>= 1)
      mx = fmaxf(mx, __shfl_xor(mx, off, 32));
    float sum = 0.0f;
    for (int n = lane; n < SEQ; n += 32) {
      const float e = __expf(row[n] - mx);
      row[n] = e;
      sum += e;
    }
    for (int off = 16; off > 0; off >>= 1) sum += __shfl_xor(sum, off, 32);
    const float inv = 1.0f / sum;
    float* arow = attn + ((size_t)bh * SEQ + qg) * SEQ;
    for (int n = lane; n < SEQ; n += 32) {
      const float p = row[n] * inv;
      row[n]  = p;
      arow[n] = p;  // attention probabilities are a required output
    }
  }
  __syncthreads();

  // ---- Phase C: ctx = P @ V ; 4 waves, one 16x16 d-slice each
  if (wave < 4) {
    const int d0 = wave * 16;
    v8f acc;
    for (int e = 0; e < 8; ++e) acc[e] = 0.0f;
    const int kEnd = q0 + 16;  // probs are exactly zero past the causal edge
    for (int k = 0; k < kEnd; k += 4) {
      const int kr = k + 2 * hi;
      v2f af = *reinterpret_cast<const v2f*>(&sc[lo * SC_STRIDE + kr]);
      v2f bf;
      bf.x = V[(size_t)(tokBase + kr) * D_MODEL + hOff + d0 + lo];
      bf.y = V[(size_t)(tokBase + kr + 1) * D_MODEL + hOff + d0 + lo];
      acc = wmma_f32(af, bf, acc);
    }
    for (int r = 0; r < 8; ++r) {
      const int m = r + 8 * hi;
      ctx[(size_t)(tokBase + q0 + m) * D_MODEL + hOff + d0 + lo] = acc[r];
    }
  }
}

// ---------------------------------------------------------------------------
extern "C" void kernel_launch(void* const* d_in, const int* in_sizes, int n_in,
                              void* d_out, int out_size, void* d_ws,
                              size_t ws_size, hipStream_t stream) {
  (void)in_sizes; (void)n_in; (void)out_size; (void)ws_size;
  const float* q  = (const float*)d_in[0];
  const float* k  = (const float*)d_in[1];
  const float* v  = (const float*)d_in[2];
  // d_in[3] is the causal mask; applied analytically inside the kernel.
  const float* Wq = (const float*)d_in[4];
  const float* bq = (const float*)d_in[5];
  const float* Wk = (const float*)d_in[6];
  const float* bk = (const float*)d_in[7];
  const float* Wv = (const float*)d_in[8];
  const float* bv = (const float*)d_in[9];
  const float* Wo = (const float*)d_in[10];
  const float* bo = (const float*)d_in[11];

  const int M = BATCH * SEQ;  // 4096 tokens
  float* out  = (float*)d_out;
  float* attn = out + (size_t)M * D_MODEL;  // outputs: (out, attn) concatenated

  float* Qb = (float*)d_ws;                  // [4096,1024] each
  float* Kb = Qb + (size_t)M * D_MODEL;
  float* Vb = Kb + (size_t)M * D_MODEL;
  float* Cb = Vb + (size_t)M * D_MODEL;

  dim3 ggrid(D_MODEL / GBN, M / GBM);  // (8, 64)
  gemm_bias_wmma<<<ggrid, 256, 0, stream>>>(q, Wq, bq, Qb, M, D_MODEL, D_MODEL);
  gemm_bias_wmma<<<ggrid, 256, 0, stream>>>(k, Wk, bk, Kb, M, D_MODEL, D_MODEL);
  gemm_bias_wmma<<<ggrid, 256, 0, stream>>>(v, Wv, bv, Vb, M, D_MODEL, D_MODEL);

  dim3 agrid(SEQ / 16, BATCH * N_HEAD);  // (128, 32)
  const size_t smem = (size_t)16 * SC_STRIDE * sizeof(float);  // 131328 B
  attn_softmax_wmma<<<agrid, 256, smem, stream>>>(Qb, Kb, Vb, attn, Cb);

  gemm_bias_wmma<<<ggrid, 256, 0, stream>>>(Cb, Wo, bo, out, M, D_MODEL, D_MODEL);
}